// StandardKVCache_43069932045032
// MI455X (gfx1250) — compile-verified
//
#include <hip/hip_runtime.h>
#include <hip/hip_bf16.h>
#include <stdint.h>

// ---- problem constants (fixed by the reference harness) ----
static constexpr int PS = 16;    // page_size
static constexpr int H  = 8;     // kv heads
static constexpr int D  = 128;   // head_dim
static constexpr int D4 = D / 4; // float4s per head row

typedef __attribute__((ext_vector_type(4))) float v4f;
typedef __attribute__((ext_vector_type(4))) int   v4i;

// pointer-to-AS1(v4i) and pointer-to-AS3(v4i), as required by the async builtins
typedef __attribute__((address_space(1))) v4i* gp4;
typedef __attribute__((address_space(3))) v4i* lp4;

// ---------------------------------------------------------------------------
// Wait helper for ASYNCcnt (gfx1250 split counters)
// ---------------------------------------------------------------------------
__device__ __forceinline__ void wait_asynccnt0() {
#if __has_builtin(__builtin_amdgcn_s_wait_asynccnt)
  __builtin_amdgcn_s_wait_asynccnt(0);
#else
  asm volatile("s_wait_asynccnt 0" ::: "memory");
#endif
}

#if __has_builtin(__builtin_amdgcn_global_load_async_to_lds_b128) && \
    __has_builtin(__builtin_amdgcn_global_store_async_from_lds_b128)
#define HAVE_ASYNC_LDS 1
#else
#define HAVE_ASYNC_LDS 0
#endif

static constexpr int COPY_THREADS = 256;
static constexpr int COPY_CHUNK   = 8;   // float4s per thread per pass

// ---------------------------------------------------------------------------
// Bulk cache copy: kv_cache (input) -> d_out, 256 MiB, pure streaming.
// Primary path: async global->LDS->global DMA (ASYNCcnt-tracked, no VGPR
// data movement). Phase-split pipeline: 8 outstanding async loads, one
// asynccnt wait, 8 async stores (drained by a second asynccnt wait each pass).
// Fallback: non-temporal b128 load/store.
// ---------------------------------------------------------------------------
__global__ __launch_bounds__(COPY_THREADS) void copy_cache_kernel(
    const v4f* __restrict__ src, v4f* __restrict__ dst, long n4) {
  const long stride = (long)gridDim.x * blockDim.x;
  const long base   = (long)blockIdx.x * blockDim.x + threadIdx.x;

#if HAVE_ASYNC_LDS
  __shared__ v4f stage[COPY_THREADS * COPY_CHUNK];
  v4f* s = const_cast<v4f*>(src);

  for (long i0 = base; i0 < n4; i0 += stride * COPY_CHUNK) {
    // phase 1: issue all async loads (global -> LDS)
#pragma unroll
    for (int j = 0; j < COPY_CHUNK; ++j) {
      const long i = i0 + (long)j * stride;
      if (i < n4) {
        __builtin_amdgcn_global_load_async_to_lds_b128(
            (gp4)(s + i), (lp4)&stage[j * COPY_THREADS + threadIdx.x],
            /*offset=*/0, /*cpol=*/0);
      }
    }
    wait_asynccnt0();  // all loads retired -> LDS slots valid
    // phase 2: issue all async stores (LDS -> global)
#pragma unroll
    for (int j = 0; j < COPY_CHUNK; ++j) {
      const long i = i0 + (long)j * stride;
      if (i < n4) {
        __builtin_amdgcn_global_store_async_from_lds_b128(
            (gp4)(dst + i), (lp4)&stage[j * COPY_THREADS + threadIdx.x],
            /*offset=*/0, /*cpol=*/0);
      }
    }
    wait_asynccnt0();  // LDS slots free for next pass
  }
#else
  for (long i = base; i < n4; i += stride) {
    v4f x = __builtin_nontemporal_load(src + i);
    __builtin_nontemporal_store(x, dst + i);
  }
#endif
}

// ---------------------------------------------------------------------------
// Ragged K/V scatter into the paged cache.
// One 256-thread block per new token. blockIdx.x (== token id) is
// wave-uniform, so all the indptr / page-table math lowers to SALU +
// scalar loads. Lane mapping: h = tid>>5, d4 = tid&31 -> each wave moves
// one contiguous 512B head row; K and V handled by the same thread.
// ---------------------------------------------------------------------------
__global__ __launch_bounds__(256) void scatter_kv_kernel(
    const v4f* __restrict__ k, const v4f* __restrict__ v,
    float* __restrict__ out,
    const int* __restrict__ append_indptr,   // [B+1]
    const int* __restrict__ page_indices,    // [B*pages_per]
    const int* __restrict__ page_indptr,     // [B+1]
    const int* __restrict__ page_lastlen,    // [B]
    int B) {
  const int t = blockIdx.x;

  // searchsorted(append_indptr, t, 'right') - 1  (B is tiny; scalar loop)
  int b = 0;
  while (b + 1 < B && append_indptr[b + 1] <= t) ++b;

  const int n_new  = append_indptr[b + 1] - append_indptr[b];
  const int local  = t - append_indptr[b];
  const int npages = page_indptr[b + 1] - page_indptr[b];
  const int seqlen = (npages - 1) * PS + page_lastlen[b];
  const int pos    = seqlen - n_new + local;
  const int page   = page_indices[page_indptr[b] + pos / PS];
  const int off    = pos - (pos / PS) * PS;

  const int tid = threadIdx.x;   // 256 = H * D4
  const int h   = tid >> 5;      // 0..7
  const int d4  = tid & 31;      // 0..31

  const long src_idx = ((long)t * H + h) * D4 + d4;
  const v4f kk = k[src_idx];
  const v4f vv = v[src_idx];

  // cache layout: [page][2][PS][H][D]
  v4f* out4 = (v4f*)out;
  const long base_k = ((((long)page * 2 + 0) * PS + off) * H + h) * D4 + d4;
  const long base_v = ((((long)page * 2 + 1) * PS + off) * H + h) * D4 + d4;
  out4[base_k] = kk;
  out4[base_v] = vv;
}

// ---------------------------------------------------------------------------
// Launch
// ---------------------------------------------------------------------------
extern "C" void kernel_launch(void* const* d_in, const int* in_sizes, int n_in,
                              void* d_out, int out_size, void* d_ws, size_t ws_size,
                              hipStream_t stream) {
  const float* k        = (const float*)d_in[0];
  const float* v        = (const float*)d_in[1];
  const float* cache_in = (const float*)d_in[2];
  const int* aind  = (const int*)d_in[3];
  const int* pidx  = (const int*)d_in[4];
  const int* pind  = (const int*)d_in[5];
  const int* plast = (const int*)d_in[6];
  float* out = (float*)d_out;

  // 1) materialize base cache into d_out (input cache -> out), async b128 DMA
  const long n4 = (long)out_size / 4;           // number of float4s
  long copy_blocks =
      (n4 + (long)COPY_THREADS * COPY_CHUNK - 1) / ((long)COPY_THREADS * COPY_CHUNK);
  if (copy_blocks < 1) copy_blocks = 1;
  copy_cache_kernel<<<(int)copy_blocks, COPY_THREADS, 0, stream>>>(
      (const v4f*)cache_in, (v4f*)out, n4);

  // 2) scatter ragged new tokens into their pages (same stream -> ordered)
  const int B = in_sizes[3] - 1;                // append_indptr has B+1 entries
  const int T = in_sizes[0] / (H * D);          // total new tokens
  scatter_kv_kernel<<<T, 256, 0, stream>>>(
      (const v4f*)k, (const v4f*)v, out, aind, pidx, pind, plast, B);
}